// DotProductSimilarity_75986561401120
// MI455X (gfx1250) — compile-verified
//
#include <hip/hip_runtime.h>
#include <hip/hip_bf16.h>
#include <math.h>

typedef __attribute__((ext_vector_type(2))) float v2f;
typedef __attribute__((ext_vector_type(8))) float v8f;

#define BATCH     1024
#define NUM_ITEMS 100000
#define DIM       256
#define TOPK      100

// ---------------------------------------------------------------------------
// GEMM: C[b,n] = sum_k A[b,k] * B[n,k]   (A=context 1024x256, B=labels 100000x256)
// Block: 256 threads = 8 waves. Block tile: 128 M x 32 N. Wave tile: 16 M x 32 N.
// B tile (32 rows x 256 K fp32 = 32KB) staged in LDS, padded stride 260 words.
// WMMA: V_WMMA_F32_16X16X4_F32, K-loop of 64 steps (K=4 each).
// ---------------------------------------------------------------------------
#define BS_STRIDE 260  // 260 mod 64 = 4 -> row-strided reads hit distinct banks

__global__ __launch_bounds__(256)
void gemm_f32_wmma(const float* __restrict__ A, const float* __restrict__ B,
                   float* __restrict__ C) {
  __shared__ float bs[32 * BS_STRIDE];

  const int tid   = threadIdx.x;
  const int nTile = blockIdx.x;          // 0..3124  (N blocks of 32)
  const int mTile = blockIdx.y;          // 0..7     (M blocks of 128)
  const int nBase = nTile * 32;
  const int mBase = mTile * 128;

  // --- cooperative load of B tile: 32 rows x 256 cols fp32 = 2048 float4 ---
  #pragma unroll
  for (int j = 0; j < 8; ++j) {
    int linear = tid + j * 256;          // float4 index, 0..2047
    int n  = linear >> 6;                // 0..31
    int c4 = (linear & 63) * 4;          // 0..252 step 4
    const float4 v = *reinterpret_cast<const float4*>(
        B + (size_t)(nBase + n) * DIM + c4);
    *reinterpret_cast<float4*>(&bs[n * BS_STRIDE + c4]) = v;
  }
  __syncthreads();

  const int wave = tid >> 5;             // 0..7 -> M strip within block
  const int lane = tid & 31;
  const int rowA = lane & 15;            // M row within 16 (also N col for B/C)
  const int half = lane >> 4;            // 0 or 1
  const int k0   = half * 2;             // K sub-offset {0,2} per A/B layout

  const float* aptr = A + (size_t)(mBase + wave * 16 + rowA) * DIM;
  const float* b0p  = &bs[(rowA)      * BS_STRIDE + k0];  // N tile 0: cols 0..15
  const float* b1p  = &bs[(rowA + 16) * BS_STRIDE + k0];  // N tile 1: cols 16..31

  v8f acc0 = {};
  v8f acc1 = {};

  #pragma unroll 8
  for (int kb = 0; kb < DIM; kb += 4) {
    v2f a  = *reinterpret_cast<const v2f*>(aptr + kb + k0);
    v2f b0 = *reinterpret_cast<const v2f*>(b0p + kb);
    v2f b1 = *reinterpret_cast<const v2f*>(b1p + kb);
    // D = A x B + C, fp32 in / fp32 accumulate (exact fp32 semantics)
    acc0 = __builtin_amdgcn_wmma_f32_16x16x4_f32(
        false, a, false, b0, (short)0, acc0, false, false);
    acc1 = __builtin_amdgcn_wmma_f32_16x16x4_f32(
        false, a, false, b1, (short)0, acc1, false, false);
  }

  // --- store: C/D layout: VGPR r, lane L -> row r + 8*half, col (L&15) ---
  const int colBase = nBase + rowA;
  #pragma unroll
  for (int r = 0; r < 8; ++r) {
    const size_t m = (size_t)(mBase + wave * 16 + r + half * 8);
    C[m * NUM_ITEMS + colBase]      = acc0[r];
    C[m * NUM_ITEMS + colBase + 16] = acc1[r];
  }
}

// ---------------------------------------------------------------------------
// Top-K: one block (256 threads) per batch row.
// 2x 256-bin radix passes on monotonic uint keys -> 16-bit threshold prefix,
// candidate collection, exact O(nc^2) rank with (value desc, index asc).
// ---------------------------------------------------------------------------
#define CAND_MAX 2048

__device__ __forceinline__ unsigned f2key(float f) {
  unsigned u = __float_as_uint(f);
  return (u & 0x80000000u) ? ~u : (u | 0x80000000u);  // larger key = larger f
}

__global__ __launch_bounds__(256)
void topk_sigmoid(const float* __restrict__ scores,
                  float* __restrict__ outIdx, float* __restrict__ outScore) {
  __shared__ unsigned hist[256];
  __shared__ unsigned s_bin1, s_above1, s_thr16;
  __shared__ float    cval[CAND_MAX];
  __shared__ int      cidx[CAND_MAX];
  __shared__ unsigned s_cnt;

  const int row = blockIdx.x;
  const int tid = threadIdx.x;
  const float* r = scores + (size_t)row * NUM_ITEMS;

  // ---- pass 1: histogram of key[31:24] ----
  hist[tid & 255] = 0;
  __syncthreads();
  for (int i = tid; i < NUM_ITEMS; i += 256)
    atomicAdd(&hist[f2key(r[i]) >> 24], 1u);
  __syncthreads();
  if (tid == 0) {
    unsigned cum = 0; int b = 255;
    for (; b > 0; --b) {
      if (cum + hist[b] >= TOPK) break;
      cum += hist[b];
    }
    s_bin1 = (unsigned)b; s_above1 = cum;
  }
  __syncthreads();
  const unsigned bin1 = s_bin1, above1 = s_above1;

  // ---- pass 2: histogram of key[23:16] restricted to bin1 ----
  hist[tid & 255] = 0;
  __syncthreads();
  for (int i = tid; i < NUM_ITEMS; i += 256) {
    unsigned k = f2key(r[i]);
    if ((k >> 24) == bin1) atomicAdd(&hist[(k >> 16) & 255u], 1u);
  }
  __syncthreads();
  if (tid == 0) {
    unsigned cum = above1; int b = 255;
    for (; b > 0; --b) {
      if (cum + hist[b] >= TOPK) break;
      cum += hist[b];
    }
    s_thr16 = (bin1 << 8) | (unsigned)b;
    s_cnt = 0;
  }
  __syncthreads();
  const unsigned thr16 = s_thr16;

  // ---- pass 3: collect candidates with key[31:16] >= thr16 ----
  for (int i = tid; i < NUM_ITEMS; i += 256) {
    float v = r[i];
    if ((f2key(v) >> 16) >= thr16) {
      unsigned p = atomicAdd(&s_cnt, 1u);
      if (p < CAND_MAX) { cval[p] = v; cidx[p] = i; }
    }
  }
  __syncthreads();
  const int nc = (int)min(s_cnt, (unsigned)CAND_MAX);

  // ---- exact rank (value desc, index asc) and emit top K ----
  for (int i = tid; i < nc; i += 256) {
    const unsigned ki = f2key(cval[i]);
    const int      ii = cidx[i];
    int rank = 0;
    for (int j = 0; j < nc; ++j) {
      unsigned kj = f2key(cval[j]);
      rank += (kj > ki) || (kj == ki && cidx[j] < ii);
    }
    if (rank < TOPK) {
      outIdx[(size_t)row * TOPK + rank]   = (float)ii;
      outScore[(size_t)row * TOPK + rank] = 1.0f / (1.0f + expf(-cval[i]));
    }
  }
}

// ---------------------------------------------------------------------------
extern "C" void kernel_launch(void* const* d_in, const int* in_sizes, int n_in,
                              void* d_out, int out_size, void* d_ws, size_t ws_size,
                              hipStream_t stream) {
  const float* ctx    = (const float*)d_in[0];   // [1024, 256]
  const float* labels = (const float*)d_in[1];   // [100000, 256]
  // d_in[2] = top_k (device scalar) -> compile-time constant 100

  float* dot      = (float*)d_out;                                   // [1024,100000]
  float* idxOut   = dot + (size_t)BATCH * NUM_ITEMS;                 // [1024,100]
  float* scoreOut = idxOut + (size_t)BATCH * TOPK;                   // [1024,100]

  dim3 gemmGrid(NUM_ITEMS / 32, BATCH / 128);     // 3125 x 8
  gemm_f32_wmma<<<gemmGrid, 256, 0, stream>>>(ctx, labels, dot);

  topk_sigmoid<<<BATCH, 256, 0, stream>>>(dot, idxOut, scoreOut);
}